// SessionReadout_24687472017536
// MI455X (gfx1250) — compile-verified
//
#include <hip/hip_runtime.h>
#include <math.h>

typedef __attribute__((ext_vector_type(2))) float v2f;
typedef __attribute__((ext_vector_type(8))) float v8f;

#define HID 256
#define NTHREADS 256

// One block per segment (batch index). Segments are contiguous because
// batch_indices is sorted, so each block:
//   1) computes scores for its rows with V_WMMA_F32_16X16X4_F32
//      (A = emb .* w premultiplied, B = ones => every D column is the score),
//   2) does the numerically-stable segment softmax in LDS,
//   3) re-reads its rows (L2-hot) for the weighted sum; thread t owns dim t.
__global__ __launch_bounds__(NTHREADS) void session_readout_fused(
    const float* __restrict__ emb,
    const int*   __restrict__ bidx,
    const float* __restrict__ att_w,
    const float* __restrict__ att_b,
    float*       __restrict__ scores,   // workspace: n_nodes floats
    float*       __restrict__ out,      // [B, HID]
    int n_nodes)
{
  const int b    = blockIdx.x;
  const int t    = threadIdx.x;
  const int wave = t >> 5;
  const int lane = t & 31;

  __shared__ float wsh[HID];       // attention weight vector
  __shared__ float red[NTHREADS];  // reduction scratch
  __shared__ float cw[NTHREADS];   // per-chunk softmax weights

  wsh[t] = att_w[t];
  __syncthreads();

  // Segment bounds via binary search (all threads redundantly; uniform).
  int lo = 0, hi = n_nodes;
  while (lo < hi) { int mid = (lo + hi) >> 1; if (bidx[mid] < b)     lo = mid + 1; else hi = mid; }
  const int segs = lo;
  hi = n_nodes;
  while (lo < hi) { int mid = (lo + hi) >> 1; if (bidx[mid] < b + 1) lo = mid + 1; else hi = mid; }
  const int sege = lo;
  const int len  = sege - segs;
  const float bias = att_b[0];

  // ---- Phase 1: scores via WMMA (16 rows per wave-group) ----
  // A layout (32-bit 16x4): lanes 0-15 hold M=0..15 {V0=K0, V1=K1};
  //                         lanes 16-31 hold M=0..15 {V0=K2, V1=K3}.
  const int ngroups = (len + 15) >> 4;
  const int koffb   = (lane >> 4) << 1;        // 0 or 2
  for (int g = wave; g < ngroups; g += (NTHREADS >> 5)) {
    const int rows_base = segs + (g << 4);
    int mrow = rows_base + (lane & 15);
    if (mrow >= sege) mrow = sege - 1;         // clamp: keeps EXEC all-ones
    const float* rowp = emb + (size_t)mrow * HID + koffb;
    v8f c = {0.f, 0.f, 0.f, 0.f, 0.f, 0.f, 0.f, 0.f};
    v2f onev; onev.x = 1.0f; onev.y = 1.0f;
    for (int k = 0; k < HID; k += 4) {
      v2f a;
      a.x = rowp[k]     * wsh[k + koffb];
      a.y = rowp[k + 1] * wsh[k + koffb + 1];
      // D = A x B + C ; B = ones => every column of D is the row-sum (score)
      c = __builtin_amdgcn_wmma_f32_16x16x4_f32(false, a, false, onev,
                                                (short)0, c, false, false);
    }
    // D layout: VGPR r, lanes 0-15 -> M=r ; lanes 16-31 -> M=8+r
    if ((lane & 15) == 0) {
      const int mb = rows_base + ((lane >> 4) << 3);
      #pragma unroll
      for (int r = 0; r < 8; ++r)
        if (mb + r < sege) scores[mb + r] = c[r] + bias;
    }
  }
  __threadfence_block();
  __syncthreads();

  // ---- Phase 2: segment softmax stats (max, then sum of exp) ----
  float mx = -INFINITY;
  for (int i = segs + t; i < sege; i += NTHREADS) mx = fmaxf(mx, scores[i]);
  red[t] = mx; __syncthreads();
  for (int off = NTHREADS >> 1; off > 0; off >>= 1) {
    if (t < off) red[t] = fmaxf(red[t], red[t + off]);
    __syncthreads();
  }
  const float segmax = red[0];
  __syncthreads();

  float sm = 0.f;
  for (int i = segs + t; i < sege; i += NTHREADS) sm += expf(scores[i] - segmax);
  red[t] = sm; __syncthreads();
  for (int off = NTHREADS >> 1; off > 0; off >>= 1) {
    if (t < off) red[t] += red[t + off];
    __syncthreads();
  }
  const float denom = red[0];
  __syncthreads();

  // ---- Phase 3: weighted accumulation; rows are L2-hot from phase 1 ----
  float acc = 0.f;
  for (int base = segs; base < sege; base += NTHREADS) {
    const int clen = min(NTHREADS, sege - base);
    if (t < clen) cw[t] = expf(scores[base + t] - segmax) / denom;
    __syncthreads();
    for (int j = 0; j < clen; ++j) {
      if (j + 1 < clen)
        __builtin_prefetch(emb + (size_t)(base + j + 1) * HID + t, 0, 1);
      acc = fmaf(cw[j], emb[(size_t)(base + j) * HID + t], acc);
    }
    __syncthreads();
  }
  out[(size_t)b * HID + t] = acc;   // empty segment -> 0 (matches reference)
}

extern "C" void kernel_launch(void* const* d_in, const int* in_sizes, int n_in,
                              void* d_out, int out_size, void* d_ws, size_t ws_size,
                              hipStream_t stream) {
  const float* emb   = (const float*)d_in[0];   // [N, 256] f32
  const int*   bidx  = (const int*)d_in[1];     // [N] sorted segment ids
  const float* att_w = (const float*)d_in[2];   // [256]
  const float* att_b = (const float*)d_in[3];   // [1]
  const int n_nodes = in_sizes[1];
  const int batch   = out_size / HID;
  float* scores = (float*)d_ws;                 // n_nodes floats of scratch

  session_readout_fused<<<batch, NTHREADS, 0, stream>>>(
      emb, bidx, att_w, att_b, scores, (float*)d_out, n_nodes);
}